// AutoSelectAttention_30889404793000
// MI455X (gfx1250) — compile-verified
//
#include <hip/hip_runtime.h>

// Gaussian banded attention scores, MI455X (gfx1250, wave32).
//
// out[b,s,l,j] = -(((j - l - L) + mean) / (var + 1e-5))^2,  L=1024, j in [0,3L)
// Flat row r = b*M + s*L + l (== span row); 3072 contiguous floats per row.
//
// Roofline: 402.7 MB written, 256 KB read -> pure streaming-store problem,
// ~17.3 us floor at 23.3 TB/s. ~4 VALU ops / 4 B element, so compute is
// ~10x under issue limits. WMMA is deliberately NOT used: the only matrix
// form (rank-3 outer product j^2 + 2cj + c^2) cancels catastrophically at
// the Gaussian center and saves nothing on a store-bound kernel.
//
// CDNA5-specific store path: compute tiles into LDS (ds_store_b128), then
// drain LDS -> HBM with GLOBAL_STORE_ASYNC_FROM_LDS_B128 (ASYNCcnt-tracked,
// CDNA5 ISA), double-buffered per wave:
//   ds_store -> s_wait_dscnt 0 -> async store -> (reuse) s_wait_asynccnt 1
// S_ENDPGM's implicit S_WAIT_IDLE covers final completion. Each lane async-
// stores only the 16 B it wrote itself, so no cross-wave barriers are needed.

#define USE_ASYNC_STORES 1

namespace {
constexpr int kL       = 1024;        // M / 4
constexpr int kThreeL  = 3 * kL;      // 3072 outputs per row
constexpr int kThreads = 256;         // 8 wave32 per block
constexpr int kVecIter = kThreeL / (4 * kThreads);  // 3 float4 tiles / thread

typedef float v4f __attribute__((ext_vector_type(4)));
}  // namespace

// ---------------------------------------------------------------------------
// CDNA5 async path: VALU -> LDS -> async DMA store to global.
// ---------------------------------------------------------------------------
__global__ __launch_bounds__(kThreads) void gauss_band_async_kernel(
    const float* __restrict__ span, float* __restrict__ out, int nrows) {
  __shared__ v4f lbuf[2][kThreads];   // 8 KB, double-buffered

  const int r = blockIdx.x;
  if (r >= nrows) return;

  // Uniform per-row parameters -> scalar loads.
  const float mean = span[2 * r + 0];
  const float var  = span[2 * r + 1];
  const float inv  = 1.0f / (var + 1e-5f);      // one divide per row
  const int   lL   = (r & (kL - 1)) + kL;       // l + L;  x_j = j - (l + L)

  float* __restrict__ orow = out + (size_t)r * (size_t)kThreeL;

  int j = (int)threadIdx.x * 4;
#pragma unroll
  for (int it = 0; it < kVecIter; ++it) {
    const float x0 = (float)(j - lL);           // exact (|x| < 2^24)
    const float x1 = x0 + 1.0f;
    const float x2 = x0 + 2.0f;
    const float x3 = x0 + 3.0f;

    const float t0 = (x0 + mean) * inv;
    const float t1 = (x1 + mean) * inv;
    const float t2 = (x2 + mean) * inv;
    const float t3 = (x3 + mean) * inv;

    v4f v;
    v.x = -(t0 * t0);
    v.y = -(t1 * t1);
    v.z = -(t2 * t2);
    v.w = -(t3 * t3);

    const int buf = it & 1;

    // Before overwriting a buffer issued 2 iterations ago, allow only the
    // most recent async store to remain outstanding (in-order completion).
    if (it >= 2) {
      asm volatile("s_wait_asynccnt 0x1" ::: "memory");
    }

    lbuf[buf][threadIdx.x] = v;                 // ds_store_b128

    // Per-lane LDS byte offset (generic LDS VA low 32 bits == LDS address).
    const unsigned lofs =
        (unsigned)(size_t)(&lbuf[buf][threadIdx.x]);
    const unsigned long long g =
        (unsigned long long)(size_t)(orow + j); // per-lane global VA

    asm volatile(
        "s_wait_dscnt 0x0\n\t"                  // LDS tile visible to DMA
        "global_store_async_from_lds_b128 %0, %1, off th:TH_STORE_NT\n\t"
        :
        : "v"(g), "v"(lofs)
        : "memory");

    j += 4 * kThreads;
  }
  // s_endpgm performs an implicit S_WAIT_IDLE -> all async stores complete.
}

// ---------------------------------------------------------------------------
// Fallback: direct b128 non-temporal stores from VGPRs (known-good).
// ---------------------------------------------------------------------------
__global__ __launch_bounds__(kThreads) void gauss_band_kernel(
    const float* __restrict__ span, float* __restrict__ out, int nrows) {
  const int r = blockIdx.x;
  if (r >= nrows) return;

  const float mean = span[2 * r + 0];
  const float var  = span[2 * r + 1];
  const float inv  = 1.0f / (var + 1e-5f);
  const int   lL   = (r & (kL - 1)) + kL;

  float* __restrict__ orow = out + (size_t)r * (size_t)kThreeL;

  int j = (int)threadIdx.x * 4;
#pragma unroll
  for (int it = 0; it < kVecIter; ++it) {
    const float x0 = (float)(j - lL);
    const float x1 = x0 + 1.0f;
    const float x2 = x0 + 2.0f;
    const float x3 = x0 + 3.0f;

    const float t0 = (x0 + mean) * inv;
    const float t1 = (x1 + mean) * inv;
    const float t2 = (x2 + mean) * inv;
    const float t3 = (x3 + mean) * inv;

    v4f v;
    v.x = -(t0 * t0);
    v.y = -(t1 * t1);
    v.z = -(t2 * t2);
    v.w = -(t3 * t3);

    __builtin_nontemporal_store(v, reinterpret_cast<v4f*>(orow + j));
    j += 4 * kThreads;
  }
}

extern "C" void kernel_launch(void* const* d_in, const int* in_sizes, int n_in,
                              void* d_out, int out_size, void* d_ws, size_t ws_size,
                              hipStream_t stream) {
  (void)n_in; (void)d_ws; (void)ws_size; (void)out_size;

  const float* span = (const float*)d_in[0];
  float*       out  = (float*)d_out;

  const int nrows = in_sizes[0] / 2;  // B*M = 32768 rows; one block per row

  dim3 grid((unsigned)nrows);
  dim3 block(kThreads);
#if USE_ASYNC_STORES
  gauss_band_async_kernel<<<grid, block, 0, stream>>>(span, out, nrows);
#else
  gauss_band_kernel<<<grid, block, 0, stream>>>(span, out, nrows);
#endif
}